// MergeLayer_67568425501389
// MI455X (gfx1250) — compile-verified
//
#include <hip/hip_runtime.h>

// CDNA5 / gfx1250. Segment-mean pooling collapsed to one weighted streaming
// reduction, with V_WMMA_F32_16X16X4_F32 as the accumulator engine.

typedef __attribute__((ext_vector_type(2))) float v2f;
typedef __attribute__((ext_vector_type(8))) float v8f;

#define N0_CAP        65536     // reference num_segments (ids clamped to this)
#define F_DIM         64
#define MAIN_BLOCKS   2048
#define BLOCK_THREADS 256

// ---------------- pass 0: zero boundary arrays ----------------
__global__ void k_zero(unsigned int* __restrict__ p, int n) {
    int i = blockIdx.x * blockDim.x + threadIdx.x;
    if (i < n) p[i] = 0u;
}

// ---------------- pass 1: segment boundaries (ids are sorted) ----------------
__global__ void k_bounds(const long long* __restrict__ seg,
                         unsigned int* __restrict__ segStart,
                         unsigned int* __restrict__ segEnd,
                         int n) {
    for (int i = blockIdx.x * blockDim.x + threadIdx.x; i < n;
         i += gridDim.x * blockDim.x) {
        int s = (int)seg[i] & (N0_CAP - 1);
        if (i == 0     || ((int)seg[i - 1] & (N0_CAP - 1)) != s) segStart[s] = (unsigned)i;
        if (i == n - 1 || ((int)seg[i + 1] & (N0_CAP - 1)) != s) segEnd[s]   = (unsigned)(i + 1);
    }
}

// ---------------- pass 2: invc[s] = 1/max(count,1), written over segStart ----------------
__global__ void k_invc(unsigned int* __restrict__ startOrInv,
                       const unsigned int* __restrict__ segEnd,
                       const int* __restrict__ pN0) {
    int n0 = *pN0; if (n0 > N0_CAP) n0 = N0_CAP;
    for (int s = blockIdx.x * blockDim.x + threadIdx.x; s < n0;
         s += gridDim.x * blockDim.x) {
        unsigned c = segEnd[s] - startOrInv[s];
        float inv = 1.0f / (float)(c ? c : 1u);
        ((float*)startOrInv)[s] = inv;   // same-thread read-then-overwrite: safe
    }
}

// ---------------- pass 2b: per-atom weight, watom[i] = invc[seg[i]] ----------------
__global__ void k_watom(const long long* __restrict__ seg,
                        const float* __restrict__ invc,
                        float* __restrict__ watom, int n) {
    for (int i = blockIdx.x * blockDim.x + threadIdx.x; i < n;
         i += gridDim.x * blockDim.x)
        watom[i] = invc[(int)seg[i] & (N0_CAP - 1)];
}

// ---------------- pass 3: weighted sum via V_WMMA_F32_16X16X4_F32 ----------------
// Per iteration a wave consumes 8 atoms (two 4-atom groups, 2 KB of x):
//   A (16x4 f32) for group gsub, tile t: A[m][k] = x[a0+4*gsub+k][16*t+m]
//     lane<16 : VGPR0 = x[..+0][16t+lane],    VGPR1 = x[..+1][16t+lane]
//     lane>=16: VGPR0 = x[..+2][16t+lane-16], VGPR1 = x[..+3][16t+lane-16]
//   B (4x16 f32): B[k][n] = w[a0+4*gsub+k] broadcast over n
//     (row-striped: VGPRv = K=v for lanes 0-15, K=v+2 for lanes 16-31)
//     -> per lane: VGPR0 = w[2*half], VGPR1 = w[2*half+1]
//   With precomputed watom this is one b64 load: {watom[i_b], watom[i_b+1]}.
//   D[m][n] = sum_k x*w (n-replicated), accumulated in C across the whole loop.
template <bool USE_WATOM>
__global__ __launch_bounds__(BLOCK_THREADS)
void k_wsum(const float* __restrict__ x, const long long* __restrict__ seg,
            const float* __restrict__ invc, const float* __restrict__ watom,
            float* __restrict__ partials, int n) {
    __shared__ float bsum[F_DIM];
    if (threadIdx.x < F_DIM) bsum[threadIdx.x] = 0.0f;
    __syncthreads();

    const int lane = threadIdx.x & 31;
    const int half = lane >> 4;      // 0: lanes 0-15, 1: lanes 16-31
    const int lmod = lane & 15;
    const int wavesPerBlock = BLOCK_THREADS / 32;
    const int nWaves = gridDim.x * wavesPerBlock;
    const int nPairs = n >> 3;       // full 8-atom iterations (tail handled below)

    const int wu = __builtin_amdgcn_readfirstlane(
        blockIdx.x * wavesPerBlock + (threadIdx.x >> 5));

    // per-lane bases: data rows and this half's weight pair
    const float*     r0 = x + ((size_t)wu * 8 + (size_t)(2 * half)) * F_DIM + lmod;
    const float*     wp = watom + (size_t)wu * 8 + (size_t)(2 * half);
    const long long* sg = seg   + (size_t)wu * 8 + (size_t)(2 * half);
    const size_t stepX = (size_t)nWaves * 8 * F_DIM;   // floats per wave-stride
    const size_t stepA = (size_t)nWaves * 8;           // atoms per wave-stride

    v8f acc[4] = {};   // one 16x16 f32 accumulator per 16-feature tile

    for (int g = wu; g < nPairs; g += nWaves) {   // wave-uniform: EXEC all-ones at WMMA
        // prefetch this wave's next iteration (global_prefetch_b8)
        __builtin_prefetch(r0 + stepX, 0, 1);

#pragma unroll
        for (int gsub = 0; gsub < 2; ++gsub) {
            v2f b;
            if (USE_WATOM) {
                b = *(const v2f*)(wp + 4 * gsub);          // one b64 broadcast load
            } else {
                int s0 = (int)sg[4 * gsub + 0] & (N0_CAP - 1);
                int s1 = (int)sg[4 * gsub + 1] & (N0_CAP - 1);
                b.x = invc[s0]; b.y = invc[s1];
            }
            const float* rg = r0 + gsub * 4 * F_DIM;       // +0 / +1024B from base
#pragma unroll
            for (int t = 0; t < 4; ++t) {
                v2f a; a.x = rg[t * 16]; a.y = rg[F_DIM + t * 16];
                acc[t] = __builtin_amdgcn_wmma_f32_16x16x4_f32(
                    false, a, false, b, (short)0, acc[t], false, false);
            }
        }
        r0 += stepX;
        wp += stepA;
        sg += stepA;
    }

    // Tail atoms (n % 8): one wave of block 0, 2 features per lane.
    if (blockIdx.x == 0 && threadIdx.x < 32) {
        for (int i = nPairs << 3; i < n; ++i) {
            float wt = invc[(int)seg[i] & (N0_CAP - 1)];
            const float* xr = x + (size_t)i * F_DIM + lane * 2;
            atomicAdd(&bsum[lane * 2 + 0], wt * xr[0]);
            atomicAdd(&bsum[lane * 2 + 1], wt * xr[1]);
        }
    }

    // D layout: VGPR v = rows M=v (lanes 0-15) / M=v+8 (lanes 16-31), n-replicated.
    // Lanes {0,16} carry all 16 values of each tile across the 8 acc VGPRs.
    if (lmod == 0) {
#pragma unroll
        for (int t = 0; t < 4; ++t)
#pragma unroll
            for (int v = 0; v < 8; ++v)
                atomicAdd(&bsum[t * 16 + v + 8 * half], acc[t][v]);  // ds_add_f32
    }
    __syncthreads();
    if (threadIdx.x < F_DIM)
        partials[(size_t)blockIdx.x * F_DIM + threadIdx.x] = bsum[threadIdx.x];
}

// ---------------- pass 4: reduce block partials, divide by N0 ----------------
__global__ void k_final(const float* __restrict__ partials, float* __restrict__ out,
                        const int* __restrict__ pN0, int nBlocks) {
    int f = threadIdx.x;
    if (f < F_DIM) {
        float s = 0.0f;
        for (int b = 0; b < nBlocks; ++b) s += partials[(size_t)b * F_DIM + f];
        int n0 = *pN0; if (n0 < 1) n0 = 1;
        out[f] = s / (float)n0;
    }
}

extern "C" void kernel_launch(void* const* d_in, const int* in_sizes, int n_in,
                              void* d_out, int out_size, void* d_ws, size_t ws_size,
                              hipStream_t stream) {
    const float*     x    = (const float*)d_in[0];
    const long long* seg  = (const long long*)d_in[1];  // int64 segment ids
    const int*       pN0  = (const int*)d_in[2];        // scalar num_segments (device)
    const int        n    = in_sizes[1];                // N atoms (in_sizes[0] = N*F)
    float*           out  = (float*)d_out;

    // ws layout: [segStart|invc : N0_CAP u32][segEnd : N0_CAP u32][partials][watom?]
    unsigned int* segStart = (unsigned int*)d_ws;
    unsigned int* segEnd   = segStart + N0_CAP;
    float*        partials = (float*)(segEnd + N0_CAP);
    float*        watom    = partials + (size_t)MAIN_BLOCKS * F_DIM;

    const size_t baseBytes  = (size_t)(2 * N0_CAP) * 4 + (size_t)MAIN_BLOCKS * F_DIM * 4;
    const bool   haveWatom  = ws_size >= baseBytes + (size_t)n * 4;

    k_zero  <<<(2 * N0_CAP + 255) / 256, 256, 0, stream>>>(segStart, 2 * N0_CAP);
    k_bounds<<<2048, 256, 0, stream>>>(seg, segStart, segEnd, n);
    k_invc  <<<(N0_CAP + 255) / 256, 256, 0, stream>>>(segStart, segEnd, pN0);

    if (haveWatom) {
        k_watom<<<2048, 256, 0, stream>>>(seg, (const float*)segStart, watom, n);
        k_wsum<true><<<MAIN_BLOCKS, BLOCK_THREADS, 0, stream>>>(
            x, seg, (const float*)segStart, watom, partials, n);
    } else {
        k_wsum<false><<<MAIN_BLOCKS, BLOCK_THREADS, 0, stream>>>(
            x, seg, (const float*)segStart, (const float*)segStart, partials, n);
    }
    k_final <<<1, 64, 0, stream>>>(partials, out, pN0, MAIN_BLOCKS);
}